// SimplePropagator_44006234915187
// MI455X (gfx1250) — compile-verified
//
#include <hip/hip_runtime.h>
#include <hip/hip_bf16.h>

typedef __attribute__((ext_vector_type(2))) float v2f;
typedef __attribute__((ext_vector_type(8))) float v8f;

#define B_ 16
#define P_ 2048
#define D_ 128
#define C3_ 384

// ---------------------------------------------------------------------------
// Kernel 1: prop[b,p,e] = sum_d z[b,p,d] * W_prop[p,d,e] + b_prop[p,e]
// One block per position p; 8 waves; wave w computes the 16x16 tile of
// columns [16w, 16w+16). fp32 WMMA 16x16x4 keeps full precision (the whole
// problem is HBM-bound on streaming fp32 weights, so fp32 matrix ops are free).
// ---------------------------------------------------------------------------
__global__ __launch_bounds__(256) void prop_kernel(
    const float* __restrict__ z, const float* __restrict__ Wp,
    const float* __restrict__ bp, float* __restrict__ prop)
{
    __shared__ __align__(16) float zt[B_ * 130];  // stride 130: bank-conflict pad
    const int p   = blockIdx.x;
    const int tid = threadIdx.x;

    // Stage z[:, p, :] (16x128 fp32 = 8KB) into LDS once, reused by all 8 waves.
    for (int i = tid; i < B_ * D_; i += 256) {
        const int m = i >> 7;        // batch row
        const int d = i & 127;
        zt[m * 130 + d] = z[(size_t)m * P_ * D_ + (size_t)p * D_ + d];
    }
    __syncthreads();

    const int lane  = tid & 31;
    const int wv    = tid >> 5;             // 0..7 -> N tile
    const int m     = lane & 15;            // A row / C col index
    const int koff  = (lane >> 4) << 1;     // 0 or 2 (K pair select)
    const int n     = (wv << 4) + m;        // output column in [0,128)

    const float* W = Wp + (size_t)p * D_ * D_;
    v8f acc = {};
#pragma unroll
    for (int k = 0; k < D_; k += 4) {
        // A fragment: 16x4 fp32, lane holds (m, k+koff), (m, k+koff+1)
        const float2 az = *(const float2*)&zt[m * 130 + k + koff];
        v2f a; a.x = az.x; a.y = az.y;
        // B fragment: 4x16 fp32, lane holds rows k+koff, k+koff+1 at col n
        v2f b;
        b.x = W[(size_t)(k + koff)     * D_ + n];
        b.y = W[(size_t)(k + koff + 1) * D_ + n];
        acc = __builtin_amdgcn_wmma_f32_16x16x4_f32(
            false, a, false, b, (short)0, acc, false, false);
    }

    const float bias = bp[(size_t)p * D_ + n];
#pragma unroll
    for (int r = 0; r < 8; ++r) {
        const int row = r + ((lane >> 4) << 3);   // C/D layout: M = r + 8*(l>=16)
        prop[(size_t)row * P_ * D_ + (size_t)p * D_ + n] = acc[r] + bias;
    }
}

// ---------------------------------------------------------------------------
// Kernel 2: out[b,p,e] = relu( sum_c cat[b,p,c] * W_vert[p,c,e] + b_vert[p,e] )
// cat = [prop[p], prop[n1], prop[n2]] assembled in LDS with edge rules:
//   n1 = (p==0) ? 1 : p-1 ;  n2 = p+1, zeroed when p==0 or p==P-1.
// ---------------------------------------------------------------------------
__global__ __launch_bounds__(256) void vert_kernel(
    const float* __restrict__ prop, const float* __restrict__ Wv,
    const float* __restrict__ bv, float* __restrict__ out)
{
    __shared__ __align__(16) float ct[B_ * 386];  // stride 386: bank-conflict pad
    const int p   = blockIdx.x;
    const int tid = threadIdx.x;

    const int  p1   = (p == 0) ? 1 : p - 1;
    const bool has2 = (p > 0) && (p < P_ - 1);
    const int  p2   = has2 ? p + 1 : 0;

    for (int i = tid; i < B_ * C3_; i += 256) {
        const int m = i / C3_;
        const int c = i - m * C3_;
        float v;
        if (c < 128)      v = prop[(size_t)m * P_ * D_ + (size_t)p  * D_ + c];
        else if (c < 256) v = prop[(size_t)m * P_ * D_ + (size_t)p1 * D_ + (c - 128)];
        else              v = has2 ? prop[(size_t)m * P_ * D_ + (size_t)p2 * D_ + (c - 256)]
                                   : 0.0f;
        ct[m * 386 + c] = v;
    }
    __syncthreads();

    const int lane = tid & 31;
    const int wv   = tid >> 5;
    const int m    = lane & 15;
    const int koff = (lane >> 4) << 1;
    const int n    = (wv << 4) + m;

    const float* W = Wv + (size_t)p * C3_ * D_;
    v8f acc = {};
#pragma unroll 8
    for (int k = 0; k < C3_; k += 4) {
        const float2 ac = *(const float2*)&ct[m * 386 + k + koff];
        v2f a; a.x = ac.x; a.y = ac.y;
        v2f b;
        b.x = W[(size_t)(k + koff)     * D_ + n];
        b.y = W[(size_t)(k + koff + 1) * D_ + n];
        acc = __builtin_amdgcn_wmma_f32_16x16x4_f32(
            false, a, false, b, (short)0, acc, false, false);
    }

    const float bias = bv[(size_t)p * D_ + n];
#pragma unroll
    for (int r = 0; r < 8; ++r) {
        const int row = r + ((lane >> 4) << 3);
        const float v = acc[r] + bias;
        out[(size_t)row * P_ * D_ + (size_t)p * D_ + n] = v > 0.0f ? v : 0.0f;
    }
}

// ---------------------------------------------------------------------------
extern "C" void kernel_launch(void* const* d_in, const int* in_sizes, int n_in,
                              void* d_out, int out_size, void* d_ws, size_t ws_size,
                              hipStream_t stream) {
    const float* z      = (const float*)d_in[0];
    const float* W_prop = (const float*)d_in[1];
    const float* b_prop = (const float*)d_in[2];
    const float* W_vert = (const float*)d_in[3];
    const float* b_vert = (const float*)d_in[4];
    float* out  = (float*)d_out;
    float* prop = (float*)d_ws;   // 16*2048*128*4 = 16.8 MB scratch

    prop_kernel<<<P_, 256, 0, stream>>>(z, W_prop, b_prop, prop);
    vert_kernel<<<P_, 256, 0, stream>>>(prop, W_vert, b_vert, out);
}